// PeriodicLinearEncodingLayer_75273596829899
// MI455X (gfx1250) — compile-verified
//
#include <hip/hip_runtime.h>

typedef __attribute__((ext_vector_type(2))) float v2f;
typedef __attribute__((ext_vector_type(4))) float v4f;
typedef __attribute__((ext_vector_type(8))) float v8f;

#define PLE_BINS 16

__global__ __launch_bounds__(256) void ple_kernel(const float* __restrict__ x,
                                                  const float* __restrict__ bnd,
                                                  float* __restrict__ out,
                                                  int n) {
    __shared__ float s_inv[PLE_BINS];
    __shared__ float s_b[PLE_BINS];

    // One lane per block sorts the 17 boundaries and derives inv = 1/(hi-lo),
    // b = -lo*inv. 17 values -> ~136 compares, negligible vs. memory time.
    if (threadIdx.x == 0) {
        float s[PLE_BINS + 1];
        #pragma unroll
        for (int i = 0; i < PLE_BINS + 1; ++i) s[i] = bnd[i];
        for (int i = 1; i < PLE_BINS + 1; ++i) {   // insertion sort
            float key = s[i];
            int j = i - 1;
            while (j >= 0 && s[j] > key) { s[j + 1] = s[j]; --j; }
            s[j + 1] = key;
        }
        for (int i = 0; i < PLE_BINS; ++i) {
            float lo = s[i];
            float inv = 1.0f / (s[i + 1] - lo);
            s_inv[i] = inv;
            s_b[i]   = -lo * inv;
        }
    }
    __syncthreads();

    const int lane   = threadIdx.x & 31;
    const int col    = lane & 15;       // N index within 16x16 tile
    const int uhalf  = lane >> 4;       // 0 = lanes 0-15, 1 = lanes 16-31
    const int wid    = blockIdx.x * (blockDim.x >> 5) + (threadIdx.x >> 5);
    const int nWaves = gridDim.x * (blockDim.x >> 5);
    const int nTiles = n >> 4;          // full 16-row tiles

    const float invc = s_inv[col];
    const float bc   = s_b[col];

#if __has_builtin(__builtin_amdgcn_wmma_f32_16x16x4_f32)
    // B (4x16): only row K=0 non-zero = inv[n]. VGPR0 lanes 0-15 hold K=0.
    v2f B;
    B.x = uhalf ? 0.0f : invc;
    B.y = 0.0f;
    // C (16x16): every row holds b[n] per column -> same value in all 8 VGPRs.
    v8f C;
    #pragma unroll
    for (int r = 0; r < 8; ++r) C[r] = bc;

    for (int tile = wid; tile < nTiles; tile += nWaves) {
        // A (16x4): only column K=0 non-zero = x[tile*16 + m].
        v2f A;
        A.x = uhalf ? 0.0f : x[tile * 16 + col];
        A.y = 0.0f;
        // D[m][n] = x[m]*inv[n] + b[n]
        v8f D = __builtin_amdgcn_wmma_f32_16x16x4_f32(
            /*neg_a=*/false, A, /*neg_b=*/false, B,
            /*c_mod=*/(short)0, C, /*reuse_a=*/false, /*reuse_b=*/false);

        // D layout: VGPR r, lanes 0-15 -> M=r; lanes 16-31 -> M=r+8, N=col.
        float* p = out + (size_t)tile * 256 + (size_t)uhalf * 128 + col;
        #pragma unroll
        for (int r = 0; r < 8; ++r) {
            float v = fminf(fmaxf(D[r], 0.0f), 1.0f);
            __builtin_nontemporal_store(v, p + r * 16);  // NT: output >> L2
        }
    }
#else
    // Fallback: one x per thread, 4x 16B NT stores (same bandwidth shape).
    const int tid    = blockIdx.x * blockDim.x + threadIdx.x;
    const int stride = gridDim.x * blockDim.x;
    for (int i = tid; i < (nTiles << 4); i += stride) {
        float xv = x[i];
        float* p = out + (size_t)i * 16;
        #pragma unroll
        for (int q = 0; q < 4; ++q) {
            v4f v;
            v.x = fminf(fmaxf(xv * s_inv[q * 4 + 0] + s_b[q * 4 + 0], 0.0f), 1.0f);
            v.y = fminf(fmaxf(xv * s_inv[q * 4 + 1] + s_b[q * 4 + 1], 0.0f), 1.0f);
            v.z = fminf(fmaxf(xv * s_inv[q * 4 + 2] + s_b[q * 4 + 2], 0.0f), 1.0f);
            v.w = fminf(fmaxf(xv * s_inv[q * 4 + 3] + s_b[q * 4 + 3], 0.0f), 1.0f);
            __builtin_nontemporal_store(v, (v4f*)(p + q * 4));
        }
    }
#endif

    // Scalar tail for n % 16 != 0 (no-op for N = 8Mi).
    const int rem = nTiles << 4;
    for (int i = rem + blockIdx.x * blockDim.x + threadIdx.x; i < n;
         i += gridDim.x * blockDim.x) {
        float xv = x[i];
        for (int j = 0; j < PLE_BINS; ++j) {
            float v = fminf(fmaxf(xv * s_inv[j] + s_b[j], 0.0f), 1.0f);
            out[(size_t)i * 16 + j] = v;
        }
    }
}

extern "C" void kernel_launch(void* const* d_in, const int* in_sizes, int n_in,
                              void* d_out, int out_size, void* d_ws, size_t ws_size,
                              hipStream_t stream) {
    const float* x   = (const float*)d_in[0];
    const float* bnd = (const float*)d_in[1];
    float* out       = (float*)d_out;
    const int n      = in_sizes[0];   // 8388608

    dim3 block(256);                  // 8 wave32 per block
    dim3 grid(1024);                  // 8192 waves grid-striding over 512K tiles
    ple_kernel<<<grid, block, 0, stream>>>(x, bnd, out, n);
}